// lruz_34857954574986
// MI455X (gfx1250) — compile-verified
//
#include <hip/hip_runtime.h>
#include <hip/hip_bf16.h>

// ---------------------------------------------------------------------------
// LRU forward pass for MI455X (gfx1250, wave32).
//  - fp32 WMMA (v_wmma_f32_16x16x4_f32) for all GEMMs: exact fp32 arithmetic,
//    matrix-pipe throughput. Problem is HBM-bound (~0.27 GB @ 23.3 TB/s),
//    so fp32 WMMA is the right precision/throughput tradeoff.
//  - chunked parallel scan (65536 chains) for the complex diagonal recurrence.
//  - setup linear algebra (spectral norms, block inverses) done on-device,
//    deterministic power iteration, single-workgroup kernels.
// ---------------------------------------------------------------------------

typedef __attribute__((ext_vector_type(2))) float v2f;
typedef __attribute__((ext_vector_type(8))) float v8f;

#define NXc   256
#define NUc   128
#define NYc   128
#define Bc    16
#define Lc    2048
#define Mrows (Bc * Lc)     /* 32768 */
#define CH    16            /* scan chunks per sequence */
#define Tch   128           /* chunk length */
#define EPSc  0.01f
#define ALPHAc 0.99f
#define PIT   120           /* power-iteration count */

// workspace offsets (floats)
static const size_t OFF_LAM   = 0;         // 512   (256 complex)
static const size_t OFF_Q     = 512;       // 256
static const size_t OFF_LAMT  = 1024;      // 512   (lam^128, complex)
static const size_t OFF_SCAL  = 1536;      // 16    [0]=dp_norm [1]=eta1 [2]=eta2 [3]=eta
static const size_t OFF_SINV  = 2048;      // 16384
static const size_t OFF_INVX4 = 18432;     // 65536
static const size_t OFF_TM    = 83968;     // 16384
static const size_t OFF_BM    = 100352;    // 32768  B  (256x128)
static const size_t OFF_CM    = 133120;    // 32768  C  (128x256)
static const size_t OFF_DM    = 165888;    // 16384  D  (128x128)
static const size_t OFF_BU    = 182272;    // 8388608  Bu (32768x256)
static const size_t OFF_SRE   = 8570880;   // 8388608  Re(states[:, :-1])
static const size_t OFF_CARRY = 16959488;  // 131072   chunk carries (complex)
static const size_t OFF_CIN   = 17090560;  // 131072   chunk carry-ins (complex)

// ---------------------------------------------------------------------------
// setup: lam, q, lam^128
__global__ void lruz_setup_lam(const float* __restrict__ nu_log,
                               const float* __restrict__ theta_log,
                               float* __restrict__ lam, float* __restrict__ q,
                               float* __restrict__ lamT) {
  int i = threadIdx.x;  // 256
  float la = expf(-expf(nu_log[i]));
  float th = expf(theta_log[i]);
  float lr = la * cosf(th), li = la * sinf(th);
  lam[2 * i] = lr; lam[2 * i + 1] = li;
  q[i] = la * la + EPSc;
  float ar = lr, ai = li;
  for (int s = 0; s < 7; ++s) { float nr = ar * ar - ai * ai, ni = 2.f * ar * ai; ar = nr; ai = ni; }
  lamT[2 * i] = ar; lamT[2 * i + 1] = ai;
}

// ---------------------------------------------------------------------------
// spectral norm of Dp (128x128) via power iteration on Dp^T Dp
__global__ void lruz_dpnorm(const float* __restrict__ Dp, float* __restrict__ scal) {
  __shared__ float v[128], w[128], red[128];
  int t = threadIdx.x;  // 128
  v[t] = 1.0f;
  __syncthreads();
  float lambda = 1.0f;
  for (int it = 0; it < PIT; ++it) {
    float acc = 0.f;
    for (int k = 0; k < 128; ++k) acc += Dp[t * 128 + k] * v[k];
    w[t] = acc;
    __syncthreads();
    float u = 0.f;
    for (int j = 0; j < 128; ++j) u += Dp[j * 128 + t] * w[j];
    red[t] = u * u;
    __syncthreads();
    for (int s = 64; s > 0; s >>= 1) { if (t < s) red[t] += red[t + s]; __syncthreads(); }
    float nrm = sqrtf(red[0]);
    lambda = nrm;
    v[t] = u / nrm;
    __syncthreads();
  }
  if (t == 0) scal[0] = sqrtf(lambda);  // sigma_max(Dp)
}

// ---------------------------------------------------------------------------
// Sinv = inv(S),  S = g*I - (1/g) M^T M,  M = c*Dp^T, c = g*alpha/dp_norm.
// S is SPD with pivots >= g(1-alpha^2) -> Gauss-Jordan without pivoting.
__global__ void lruz_invS(const float* __restrict__ Dp, const float* __restrict__ scal,
                          const float* __restrict__ gamma_p, float* __restrict__ Sinv) {
  extern __shared__ float S[];  // 128*128 floats
  int t = threadIdx.x;  // 128
  float g = gamma_p[0], dpn = scal[0];
  float c = ALPHAc * g / dpn;
  for (int j = 0; j < 128; ++j) {
    float acc = 0.f;
    for (int k = 0; k < 128; ++k) acc += Dp[t * 128 + k] * Dp[j * 128 + k];
    S[t * 128 + j] = ((t == j) ? g : 0.f) - (c * c / g) * acc;
    Sinv[t * 128 + j] = (t == j) ? 1.f : 0.f;
  }
  __syncthreads();
  for (int p = 0; p < 128; ++p) {
    float piv = S[p * 128 + p];
    float f = S[t * 128 + p] / piv;
    if (t != p) {
      for (int j = 0; j < 128; ++j) {
        S[t * 128 + j]    -= f * S[p * 128 + j];
        Sinv[t * 128 + j] -= f * Sinv[p * 128 + j];
      }
    }
    __syncthreads();
    if (t == p) {
      float inv = 1.f / piv;
      for (int j = 0; j < 128; ++j) { S[p * 128 + j] *= inv; Sinv[p * 128 + j] *= inv; }
    }
    __syncthreads();
  }
}

// Tm = M * Sinv   (Tm[a][j] = c * sum_b Dp[b][a]*Sinv[b][j])
__global__ void lruz_Tm(const float* __restrict__ Dp, const float* __restrict__ Sinv,
                        const float* __restrict__ scal, const float* __restrict__ gamma_p,
                        float* __restrict__ Tm) {
  int a = blockIdx.x, j = threadIdx.x;
  float c = ALPHAc * gamma_p[0] / scal[0];
  float acc = 0.f;
  for (int b = 0; b < 128; ++b) acc += Dp[b * 128 + a] * Sinv[b * 128 + j];
  Tm[a * 128 + j] = c * acc;
}

// assemble symmetric inv(X4) (256x256) from Schur blocks
__global__ void lruz_invX4(const float* __restrict__ Dp, const float* __restrict__ Sinv,
                           const float* __restrict__ Tm, const float* __restrict__ scal,
                           const float* __restrict__ gamma_p, float* __restrict__ invX4) {
  int i = blockIdx.x, j = threadIdx.x;  // 256 x 256
  float g = gamma_p[0];
  float c = ALPHAc * g / scal[0];
  float val;
  if (i < 128 && j < 128) {
    float acc = 0.f;
    for (int k = 0; k < 128; ++k) acc += Tm[i * 128 + k] * Dp[k * 128 + j];
    val = ((i == j) ? 1.f / g : 0.f) + (c / (g * g)) * acc;
  } else if (i < 128) {
    val = -Tm[i * 128 + (j - 128)] / g;
  } else if (j < 128) {
    val = -Tm[j * 128 + (i - 128)] / g;
  } else {
    val = Sinv[(i - 128) * 128 + (j - 128)];
  }
  invX4[i * 256 + j] = val;
}

// ---------------------------------------------------------------------------
// eta1 = ||inv(X1) X2t||_2 : power iteration; inv(X1) = per-index 2x2 complex
// Hermitian solve: a = (y1 - lam*y2)/d, b = (y2 - conj(lam)*y1)/d, d=q(1-|lam|^2)
__global__ void lruz_eta1(const float* __restrict__ X2b, const float* __restrict__ lam,
                          const float* __restrict__ q, float* __restrict__ scal) {
  __shared__ float2 v[256], w[512];
  __shared__ float red[256];
  int t = threadIdx.x;  // 256
  v[t] = make_float2(1.f, 0.f);
  __syncthreads();
  float lambda = 1.f;
  float lr = lam[2 * t], li = lam[2 * t + 1];
  float d = q[t] * (1.f - (lr * lr + li * li));
  for (int it = 0; it < PIT; ++it) {
    // y = X2t v  (rows t and t+256; mask zeroes off-diagonal blocks)
    float yr = 0.f, yi = 0.f, zr = 0.f, zi = 0.f;
    for (int c = 0; c < 128; ++c) {
      float x = X2b[t * 256 + c];
      yr += x * v[c].x; yi += x * v[c].y;
    }
    for (int c = 128; c < 256; ++c) {
      float x = X2b[(t + 256) * 256 + c];
      zr += x * v[c].x; zi += x * v[c].y;
    }
    // w = inv(X1) y  (pairwise)
    float ar = (yr - (lr * zr - li * zi)) / d;
    float ai = (yi - (lr * zi + li * zr)) / d;
    float br = (zr - (lr * yr + li * yi)) / d;
    float bi = (zi - (lr * yi - li * yr)) / d;
    w[t] = make_float2(ar, ai);
    w[t + 256] = make_float2(br, bi);
    __syncthreads();
    // u = X2t^T (inv(X1) w) -- inv(X1) Hermitian; second apply folded below:
    // since G = M^H M with M = invX1*X2t and invX1^H = invX1, u = X2t^T (invX1 * invX1 * y)
    // we instead follow M^H(Mv): w already = Mv; apply invX1 again then X2t^T.
    float p1r = (w[t].x - (lr * w[t + 256].x - li * w[t + 256].y)) / d;
    float p1i = (w[t].y - (lr * w[t + 256].y + li * w[t + 256].x)) / d;
    float p2r = (w[t + 256].x - (lr * w[t].x + li * w[t].y)) / d;
    float p2i = (w[t + 256].y - (lr * w[t].y - li * w[t].x)) / d;
    __syncthreads();
    w[t] = make_float2(p1r, p1i);
    w[t + 256] = make_float2(p2r, p2i);
    __syncthreads();
    float ur = 0.f, ui = 0.f;
    if (t < 128) {
      for (int r = 0; r < 256; ++r) { float x = X2b[r * 256 + t]; ur += x * w[r].x; ui += x * w[r].y; }
    } else {
      for (int r = 256; r < 512; ++r) { float x = X2b[r * 256 + t]; ur += x * w[r].x; ui += x * w[r].y; }
    }
    red[t] = ur * ur + ui * ui;
    __syncthreads();
    for (int s = 128; s > 0; s >>= 1) { if (t < s) red[t] += red[t + s]; __syncthreads(); }
    float nrm = sqrtf(red[0]);
    lambda = nrm;
    v[t] = make_float2(ur / nrm, ui / nrm);
    __syncthreads();
  }
  if (t == 0) scal[1] = sqrtf(lambda);
}

// eta2 = ||X2t inv(X4)||_2 : power iteration on invX4^T X2t^T X2t invX4
__global__ void lruz_eta2(const float* __restrict__ X2b, const float* __restrict__ invX4,
                          float* __restrict__ scal) {
  __shared__ float v[256], tv[256], w[512], u[256], red[256];
  int t = threadIdx.x;  // 256
  v[t] = 1.f;
  __syncthreads();
  float lambda = 1.f;
  for (int it = 0; it < PIT; ++it) {
    float a = 0.f;
    for (int j = 0; j < 256; ++j) a += invX4[t * 256 + j] * v[j];
    tv[t] = a;
    __syncthreads();
    float w1 = 0.f, w2 = 0.f;
    for (int c = 0; c < 128; ++c)   w1 += X2b[t * 256 + c] * tv[c];
    for (int c = 128; c < 256; ++c) w2 += X2b[(t + 256) * 256 + c] * tv[c];
    w[t] = w1; w[t + 256] = w2;
    __syncthreads();
    float uu = 0.f;
    if (t < 128) { for (int r = 0; r < 256; ++r)   uu += X2b[r * 256 + t] * w[r]; }
    else         { for (int r = 256; r < 512; ++r) uu += X2b[r * 256 + t] * w[r]; }
    u[t] = uu;
    __syncthreads();
    float b = 0.f;
    for (int j = 0; j < 256; ++j) b += invX4[t * 256 + j] * u[j];  // symmetric
    red[t] = b * b;
    __syncthreads();
    for (int s = 128; s > 0; s >>= 1) { if (t < s) red[t] += red[t + s]; __syncthreads(); }
    float nrm = sqrtf(red[0]);
    lambda = nrm;
    v[t] = b / nrm;
    __syncthreads();
  }
  if (t == 0) scal[2] = sqrtf(lambda);
}

// finalize: eta, B (256x128), C (128x256), D (128x128) -- all real fp32
__global__ void lruz_finalize(const float* __restrict__ X2b, const float* __restrict__ Dp,
                              const float* __restrict__ q, const float* __restrict__ gamma_p,
                              float* __restrict__ scal, float* __restrict__ Bm,
                              float* __restrict__ Cm, float* __restrict__ Dm) {
  int n = blockIdx.x, t = threadIdx.x;  // 256 x 128
  float e = fmaxf(fmaxf(scal[1], scal[2]), 1.0f);
  if (n == 0 && t == 0) scal[3] = e;
  Bm[n * 128 + t] = X2b[n * 256 + t] / (e * q[n]);
  Cm[t * 256 + n] = X2b[(256 + n) * 256 + 128 + t] / e;
  if (n < 128) Dm[n * 128 + t] = (gamma_p[0] * ALPHAc / scal[0]) * Dp[n * 128 + t];
}

// ---------------------------------------------------------------------------
// GEMM1: Bu (32768x256) = X (32768x128) @ Bm^T   -- fp32 WMMA 16x16x4
__global__ void lruz_gemm1(const float* __restrict__ X, const float* __restrict__ Bm,
                           float* __restrict__ Bu) {
  const int K = 128, N = 256;
  int wave = threadIdx.x >> 5;       // 8 waves; each covers 32 cols (2 tiles)
  int lane = threadIdx.x & 31;
  int m0 = blockIdx.x * 16;
  int n0 = wave * 32;
  int mrow = lane & 15;
  int kb = (lane >> 4) * 2;          // A/B frag k sub-offset per ISA layout
  v8f acc0 = {}; v8f acc1 = {};
  const float* Arow = X + (size_t)(m0 + mrow) * K;
  const float* B0 = Bm + (size_t)(n0 + mrow) * K;
  const float* B1 = Bm + (size_t)(n0 + 16 + mrow) * K;
#pragma unroll 4
  for (int k = 0; k < K; k += 4) {
    __builtin_prefetch(Arow + k + 64, 0, 0);
    v2f a  = *(const v2f*)(Arow + k + kb);
    v2f b0 = *(const v2f*)(B0 + k + kb);
    v2f b1 = *(const v2f*)(B1 + k + kb);
    acc0 = __builtin_amdgcn_wmma_f32_16x16x4_f32(false, a, false, b0, (short)0, acc0, false, false);
    acc1 = __builtin_amdgcn_wmma_f32_16x16x4_f32(false, a, false, b1, (short)0, acc1, false, false);
  }
  int ncol = lane & 15;
  int mhi = (lane >> 4) * 8;
#pragma unroll
  for (int r = 0; r < 8; ++r) {
    Bu[(size_t)(m0 + mhi + r) * N + n0 + ncol]      = acc0[r];
    Bu[(size_t)(m0 + mhi + r) * N + n0 + 16 + ncol] = acc1[r];
  }
}

// ---------------------------------------------------------------------------
// scan phase 1: per-(batch,chunk,state) local scan -> carry (complex)
__global__ void lruz_scan_carry(const float* __restrict__ Bu, const float* __restrict__ lam,
                                float* __restrict__ carry) {
  int tid = blockIdx.x * blockDim.x + threadIdx.x;  // 65536
  int n = tid & 255;
  int c = (tid >> 8) & (CH - 1);
  int b = tid >> 12;
  float lr = lam[2 * n], li = lam[2 * n + 1];
  float sr = 0.f, si = 0.f;
  const float* bu = Bu + ((size_t)b * Lc + (size_t)c * Tch) * NXc + n;
  for (int t = 0; t < Tch; ++t) {
    float u = bu[(size_t)t * NXc];
    float nr = lr * sr - li * si + u;
    float ni = lr * si + li * sr;
    sr = nr; si = ni;
  }
  size_t idx = ((size_t)b * CH + c) * NXc + n;
  carry[2 * idx] = sr; carry[2 * idx + 1] = si;
}

// scan phase 2: combine chunk carries with lam^128 -> per-chunk carry-in
__global__ void lruz_scan_combine(const float* __restrict__ carry, const float* __restrict__ lamT,
                                  float* __restrict__ cin) {
  int tid = blockIdx.x * blockDim.x + threadIdx.x;  // 4096
  int n = tid & 255;
  int b = tid >> 8;
  float Tr = lamT[2 * n], Ti = lamT[2 * n + 1];
  float sr = 0.f, si = 0.f;
  for (int c = 0; c < CH; ++c) {
    size_t idx = ((size_t)b * CH + c) * NXc + n;
    cin[2 * idx] = sr; cin[2 * idx + 1] = si;
    float cr = carry[2 * idx], ci = carry[2 * idx + 1];
    float nr = Tr * sr - Ti * si + cr;
    float ni = Tr * si + Ti * sr + ci;
    sr = nr; si = ni;
  }
}

// scan phase 3: emit complex states (d_out) and real part (Sre) for GEMM2
__global__ void lruz_scan_emit(const float* __restrict__ Bu, const float* __restrict__ lam,
                               const float* __restrict__ cin, float* __restrict__ states,
                               float* __restrict__ Sre) {
  int tid = blockIdx.x * blockDim.x + threadIdx.x;  // 65536
  int n = tid & 255;
  int c = (tid >> 8) & (CH - 1);
  int b = tid >> 12;
  float lr = lam[2 * n], li = lam[2 * n + 1];
  size_t cidx = ((size_t)b * CH + c) * NXc + n;
  float sr = cin[2 * cidx], si = cin[2 * cidx + 1];
  if (c == 0) {
    size_t s0 = ((size_t)b * (Lc + 1)) * NXc + n;
    states[2 * s0] = 0.f; states[2 * s0 + 1] = 0.f;   // x0 = 0
    Sre[(size_t)b * Lc * NXc + n] = 0.f;
  }
  const float* bu = Bu + ((size_t)b * Lc + (size_t)c * Tch) * NXc + n;
  for (int t = 0; t < Tch; ++t) {
    int tt = c * Tch + t;
    float u = bu[(size_t)t * NXc];
    float nr = lr * sr - li * si + u;
    float ni = lr * si + li * sr;
    sr = nr; si = ni;
    size_t sidx = ((size_t)b * (Lc + 1) + tt + 1) * NXc + n;
    states[2 * sidx] = sr; states[2 * sidx + 1] = si;
    if (tt + 1 < Lc) Sre[((size_t)b * Lc + tt + 1) * NXc + n] = sr;
  }
}

// ---------------------------------------------------------------------------
// GEMM2 (fused): Out (32768x128) = Sre(32768x256)@Cm^T + X(32768x128)@Dm^T
__global__ void lruz_gemm2(const float* __restrict__ Sre, const float* __restrict__ X,
                           const float* __restrict__ Cm, const float* __restrict__ Dm,
                           float* __restrict__ Out) {
  int wave = threadIdx.x >> 5;       // 8 waves; each one 16-col tile
  int lane = threadIdx.x & 31;
  int m0 = blockIdx.x * 16;
  int n0 = wave * 16;
  int mrow = lane & 15;
  int kb = (lane >> 4) * 2;
  v8f acc = {};
  const float* A1 = Sre + (size_t)(m0 + mrow) * 256;
  const float* B1 = Cm + (size_t)(n0 + mrow) * 256;
#pragma unroll 4
  for (int k = 0; k < 256; k += 4) {
    __builtin_prefetch(A1 + k + 64, 0, 0);
    v2f a = *(const v2f*)(A1 + k + kb);
    v2f b = *(const v2f*)(B1 + k + kb);
    acc = __builtin_amdgcn_wmma_f32_16x16x4_f32(false, a, false, b, (short)0, acc, false, false);
  }
  const float* A2 = X + (size_t)(m0 + mrow) * 128;
  const float* B2 = Dm + (size_t)(n0 + mrow) * 128;
#pragma unroll 4
  for (int k = 0; k < 128; k += 4) {
    v2f a = *(const v2f*)(A2 + k + kb);
    v2f b = *(const v2f*)(B2 + k + kb);
    acc = __builtin_amdgcn_wmma_f32_16x16x4_f32(false, a, false, b, (short)0, acc, false, false);
  }
  int ncol = lane & 15, mhi = (lane >> 4) * 8;
#pragma unroll
  for (int r = 0; r < 8; ++r)
    Out[(size_t)(m0 + mhi + r) * 128 + n0 + ncol] = acc[r];
}

// ---------------------------------------------------------------------------
extern "C" void kernel_launch(void* const* d_in, const int* in_sizes, int n_in,
                              void* d_out, int out_size, void* d_ws, size_t ws_size,
                              hipStream_t stream) {
  (void)in_sizes; (void)n_in; (void)out_size; (void)ws_size;
  const float* X        = (const float*)d_in[0];  // (16,2048,128)
  const float* nu_log   = (const float*)d_in[1];  // (256,)
  const float* theta_log= (const float*)d_in[2];  // (256,)
  const float* gamma_p  = (const float*)d_in[3];  // scalar
  const float* X2b      = (const float*)d_in[4];  // (512,256)
  const float* Dp       = (const float*)d_in[5];  // (128,128)

  float* out = (float*)d_out;                     // [output | states(complex)]
  float* states = out + (size_t)Bc * Lc * NYc;    // complex interleaved

  float* ws   = (float*)d_ws;
  float* lam  = ws + OFF_LAM;
  float* q    = ws + OFF_Q;
  float* lamT = ws + OFF_LAMT;
  float* scal = ws + OFF_SCAL;
  float* Sinv = ws + OFF_SINV;
  float* invX4= ws + OFF_INVX4;
  float* Tm   = ws + OFF_TM;
  float* Bm   = ws + OFF_BM;
  float* Cm   = ws + OFF_CM;
  float* Dm   = ws + OFF_DM;
  float* Bu   = ws + OFF_BU;
  float* Sre  = ws + OFF_SRE;
  float* carry= ws + OFF_CARRY;
  float* cin  = ws + OFF_CIN;

  // --- parameter setup (tiny kernels) ---
  lruz_setup_lam<<<1, 256, 0, stream>>>(nu_log, theta_log, lam, q, lamT);
  lruz_dpnorm<<<1, 128, 0, stream>>>(Dp, scal);
  lruz_invS<<<1, 128, 128 * 128 * sizeof(float), stream>>>(Dp, scal, gamma_p, Sinv);
  lruz_Tm<<<128, 128, 0, stream>>>(Dp, Sinv, scal, gamma_p, Tm);
  lruz_invX4<<<256, 256, 0, stream>>>(Dp, Sinv, Tm, scal, gamma_p, invX4);
  lruz_eta1<<<1, 256, 0, stream>>>(X2b, lam, q, scal);
  lruz_eta2<<<1, 256, 0, stream>>>(X2b, invX4, scal);
  lruz_finalize<<<256, 128, 0, stream>>>(X2b, Dp, q, gamma_p, scal, Bm, Cm, Dm);

  // --- heavy phase ---
  lruz_gemm1<<<Mrows / 16, 256, 0, stream>>>(X, Bm, Bu);
  lruz_scan_carry<<<256, 256, 0, stream>>>(Bu, lam, carry);
  lruz_scan_combine<<<16, 256, 0, stream>>>(carry, lamT, cin);
  lruz_scan_emit<<<256, 256, 0, stream>>>(Bu, lam, cin, states, Sre);
  lruz_gemm2<<<Mrows / 16, 256, 0, stream>>>(Sre, X, Cm, Dm, out);
}